// InvertedDotProductAttention_27857157882328
// MI455X (gfx1250) — compile-verified
//
#include <hip/hip_runtime.h>
#include <cstdint>

typedef __attribute__((ext_vector_type(2))) float v2f;
typedef __attribute__((ext_vector_type(4))) float v4f;
typedef __attribute__((ext_vector_type(8))) float v8f;
typedef int v4i_ __attribute__((vector_size(16)));  // matches builtin proto

#define AS1 __attribute__((address_space(1)))
#define AS3 __attribute__((address_space(3)))

namespace {
constexpr int B_ = 8, Q_ = 16, K_ = 65536, D_ = 64;
constexpr int WAVES = 8, THREADS = WAVES * 32;
constexpr int TILES_PER_WAVE = 8;
constexpr int KB = WAVES * TILES_PER_WAVE * 16;   // 1024 keys per block
constexpr int KBLOCKS = K_ / KB;                  // 64
constexpr int KPAD = 68;   // LDS row stride (floats) for key tile -> conflict-free
constexpr int APAD = 17;   // LDS row stride (floats) for attn tile
constexpr float EPS = 1e-8f;
constexpr float QSCALE = 0.125f;  // 1/sqrt(64)
}

#if __has_builtin(__builtin_amdgcn_global_load_async_to_lds_b128)
#define HAS_ASYNC_LDS 1
#else
#define HAS_ASYNC_LDS 0
#endif

#if __has_builtin(__builtin_amdgcn_s_wait_asynccnt)
#define WAIT_ASYNCCNT(n)                    \
  do {                                      \
    __builtin_amdgcn_s_wait_asynccnt(n);    \
    asm volatile("" ::: "memory");          \
  } while (0)
#else
#define WAIT_ASYNCCNT(n) asm volatile("s_wait_asynccnt %0" ::"i"(n) : "memory")
#endif

__device__ __forceinline__ v8f zero8() {
  v8f v;
#pragma unroll
  for (int i = 0; i < 8; ++i) v[i] = 0.0f;
  return v;
}

// one 16-byte async global->LDS copy (ASYNCcnt-tracked on gfx1250)
__device__ __forceinline__ void async_cp16(const float* g, float* l) {
#if HAS_ASYNC_LDS
  __builtin_amdgcn_global_load_async_to_lds_b128(
      (AS1 v4i_*)(uintptr_t)g, (AS3 v4i_*)(uint32_t)(uintptr_t)l, 0, 0);
#endif
}

// stage one contiguous 16x64 f32 key tile into bank-padded LDS rows
__device__ __forceinline__ void stage_key_async(const float* gsrc, float* lbuf,
                                                int lane) {
#pragma unroll
  for (int i = 0; i < 8; ++i) {
    const int e = (i * 32 + lane) * 4;
    async_cp16(gsrc + e, lbuf + (e >> 6) * KPAD + (e & 63));
  }
}

__device__ __forceinline__ void stage_key_sync(const float* gsrc, float* lbuf,
                                               int lane) {
#pragma unroll
  for (int i = 0; i < 8; ++i) {
    const int e = (i * 32 + lane) * 4;
    const v4f kv = *(const v4f*)(gsrc + e);
    *(v4f*)(lbuf + (e >> 6) * KPAD + (e & 63)) = kv;
  }
}

// ---------------------------------------------------------------------------
// Kernel 0: zero the workspace accumulators (norm[128] + acc[8192])
// ---------------------------------------------------------------------------
__global__ void iattn_zero_ws(float* __restrict__ ws, int n) {
  int i = blockIdx.x * blockDim.x + threadIdx.x;
  if (i < n) ws[i] = 0.0f;
}

// ---------------------------------------------------------------------------
// Kernel 1: fused scores -> softmax(Q-axis) -> attn store -> rowsum -> attn*V
// One wave owns 16-key tiles; key stream double-buffered in LDS via async
// global->LDS copies; 32 V_WMMA_F32_16X16X4_F32 per tile.
// ---------------------------------------------------------------------------
__global__ __launch_bounds__(THREADS) void iattn_main(
    const float* __restrict__ qry, const float* __restrict__ key,
    const float* __restrict__ val, float* __restrict__ attn_out,
    float* __restrict__ norm_acc, float* __restrict__ out_acc) {
  __shared__ float lds_key[WAVES][2][16 * KPAD];  // 69.6 KB, double-buffered
  __shared__ float lds_attn[WAVES][16 * APAD];    // 8.7 KB, transpose bounce
  __shared__ float lds_oacc[Q_ * D_];             // 4 KB block accumulator

  const int b    = blockIdx.y;
  const int blk  = blockIdx.x;
  const int tid  = threadIdx.x;
  const int wave = tid >> 5;
  const int lane = tid & 31;
  const int ln16 = lane & 15;
  const int hf   = lane >> 4;  // which 16-lane half

  for (int i = tid; i < Q_ * D_; i += THREADS) lds_oacc[i] = 0.0f;
  __syncthreads();

  // ---- load Q once, pre-scaled, as WMMA A-fragments (16x4 f32 layout)
  const float* qb = qry + (size_t)b * Q_ * D_;
  v2f qa[16];
#pragma unroll
  for (int f = 0; f < 16; ++f) {
    const int d0 = 4 * f + 2 * hf;
    qa[f].x = QSCALE * qb[ln16 * D_ + d0];
    qa[f].y = QSCALE * qb[ln16 * D_ + d0 + 1];
  }

  v8f oacc[4];  // 16x64 attn*V accumulator (4 independent 16x16 chains)
#pragma unroll
  for (int n = 0; n < 4; ++n) oacc[n] = zero8();
  float racc[8];
#pragma unroll
  for (int g = 0; g < 8; ++g) racc[g] = 0.0f;

  const float* keyb = key + (size_t)b * K_ * D_;
  const float* valb = val + (size_t)b * K_ * D_;
  float* lattn = lds_attn[wave];

#if HAS_ASYNC_LDS
  // prologue: async-stage tile 0
  stage_key_async(keyb + (size_t)(blk * KB + wave * 16) * D_,
                  lds_key[wave][0], lane);
#endif

  for (int it = 0; it < TILES_PER_WAVE; ++it) {
    const int kbase = blk * KB + (it * WAVES + wave) * 16;
    const float* vsrc = valb + (size_t)kbase * D_;
    float* lkey = lds_key[wave][it & 1];

#if HAS_ASYNC_LDS
    if (it + 1 < TILES_PER_WAVE) {
      // issue next tile into the other buffer, then wait for current tile:
      // <=8 outstanding async ops  ==>  this tile's 8 copies have landed.
      stage_key_async(keyb + (size_t)(kbase + WAVES * 16) * D_,
                      lds_key[wave][(it + 1) & 1], lane);
      WAIT_ASYNCCNT(8);
    } else {
      WAIT_ASYNCCNT(0);
    }
#else
    stage_key_sync(keyb + (size_t)kbase * D_, lkey, lane);
    asm volatile("s_wait_dscnt 0" ::: "memory");
#endif
    if (it + 1 < TILES_PER_WAVE)
      __builtin_prefetch(vsrc + (size_t)WAVES * 16 * D_ + lane * 32, 0, 0);

    // ---- scores C[16x16] = (Q*scale) x key^T : 16 WMMAs over D=64,
    //      two interleaved accumulator chains for ILP
    v8f c0 = zero8(), c1 = zero8();
#pragma unroll
    for (int f = 0; f < 8; ++f) {
      const int d0a = 8 * f + 2 * hf;
      const int d0b = 8 * f + 4 + 2 * hf;
      v2f ba, bb;  // B 4x16: lane = column n, rows 2*half+{0,1}
      ba.x = lkey[ln16 * KPAD + d0a];
      ba.y = lkey[ln16 * KPAD + d0a + 1];
      bb.x = lkey[ln16 * KPAD + d0b];
      bb.y = lkey[ln16 * KPAD + d0b + 1];
      c0 = __builtin_amdgcn_wmma_f32_16x16x4_f32(false, qa[2 * f], false, ba,
                                                 (short)0, c0, false, false);
      c1 = __builtin_amdgcn_wmma_f32_16x16x4_f32(false, qa[2 * f + 1], false,
                                                 bb, (short)0, c1, false,
                                                 false);
    }
    const v8f c = c0 + c1;

    // ---- softmax over the 16 queries (column of the tile):
    // lane L holds rows g+8*half of column ln16; partner lane is L^16.
    float mx = c[0];
#pragma unroll
    for (int g = 1; g < 8; ++g) mx = fmaxf(mx, c[g]);
    mx = fmaxf(mx, __shfl_xor(mx, 16, 32));
    float ex[8];
    float sm = 0.0f;
#pragma unroll
    for (int g = 0; g < 8; ++g) {
      ex[g] = __expf(c[g] - mx);
      sm += ex[g];
    }
    sm += __shfl_xor(sm, 16, 32);
    const float inv = __frcp_rn(sm);
#pragma unroll
    for (int g = 0; g < 8; ++g) ex[g] *= inv;

    // ---- emit attn tile (output #2), stash transpose copy, track row sums
    float* aout = attn_out + (size_t)b * Q_ * K_;
#pragma unroll
    for (int g = 0; g < 8; ++g) {
      const int qrow = g + 8 * hf;
      aout[(size_t)qrow * K_ + kbase + ln16] = ex[g];  // coalesced over k
      lattn[qrow * APAD + ln16] = ex[g];
      racc[g] += ex[g];
    }
    asm volatile("s_wait_dscnt 0" ::: "memory");

    // ---- accumulate attn[16x16] x value[16x64]: 16 WMMAs, 4 indep chains
#pragma unroll
    for (int j = 0; j < 4; ++j) {  // 4-key chunk within tile
      const int kk0 = 4 * j + 2 * hf;
      v2f af;  // A 16x4: lane row m=ln16, cols kk0+{0,1}
      af.x = lattn[ln16 * APAD + kk0];
      af.y = lattn[ln16 * APAD + kk0 + 1];
#pragma unroll
      for (int n = 0; n < 4; ++n) {  // 16-dim chunk of D=64
        const float* vrow = vsrc + (size_t)kk0 * D_ + n * 16 + ln16;
        v2f bf;  // B 4x16: rows kk0,{+1}; column dim = n*16+ln16
        bf.x = vrow[0];
        bf.y = vrow[D_];
        oacc[n] = __builtin_amdgcn_wmma_f32_16x16x4_f32(
            false, af, false, bf, (short)0, oacc[n], false, false);
      }
    }
  }

  // ---- normalizer: reduce racc over the 16 lanes of each half, one atomic
#pragma unroll
  for (int g = 0; g < 8; ++g) {
    float r = racc[g];
    r += __shfl_xor(r, 1, 32);
    r += __shfl_xor(r, 2, 32);
    r += __shfl_xor(r, 4, 32);
    r += __shfl_xor(r, 8, 32);
    if (ln16 == 0) atomicAdd(&norm_acc[b * Q_ + g + 8 * hf], r);
  }

  // ---- merge wave accumulators in LDS (ds_add_f32), then one block flush
#pragma unroll
  for (int n = 0; n < 4; ++n)
#pragma unroll
    for (int g = 0; g < 8; ++g)
      atomicAdd(&lds_oacc[(g + 8 * hf) * D_ + n * 16 + ln16], oacc[n][g]);
  __syncthreads();
  for (int i = tid; i < Q_ * D_; i += THREADS)
    atomicAdd(&out_acc[(size_t)b * Q_ * D_ + i], lds_oacc[i]);
}

// ---------------------------------------------------------------------------
// Kernel 2: output[b,q,d] = acc[b,q,d] / (norm[b,q] + EPS)
// ---------------------------------------------------------------------------
__global__ void iattn_finalize(const float* __restrict__ acc,
                               const float* __restrict__ norm,
                               float* __restrict__ out) {
  const int i = blockIdx.x * blockDim.x + threadIdx.x;
  if (i < B_ * Q_ * D_) {
    const int bq = i / D_;  // b*16 + q
    out[i] = acc[i] / (norm[bq] + EPS);
  }
}

extern "C" void kernel_launch(void* const* d_in, const int* in_sizes, int n_in,
                              void* d_out, int out_size, void* d_ws,
                              size_t ws_size, hipStream_t stream) {
  (void)in_sizes; (void)n_in; (void)out_size; (void)ws_size;
  const float* qry = (const float*)d_in[0];
  const float* key = (const float*)d_in[1];
  const float* val = (const float*)d_in[2];

  float* out  = (float*)d_out;            // [B,Q,D]   : 8192 floats
  float* attn = out + B_ * Q_ * D_;       // [B,H,Q,K] : 8.4M floats
  float* norm = (float*)d_ws;             // [B,Q]     : 128 floats
  float* acc  = norm + B_ * Q_;           // [B,Q,D]   : 8192 floats

  const int nws = B_ * Q_ + B_ * Q_ * D_;
  iattn_zero_ws<<<(nws + 255) / 256, 256, 0, stream>>>(norm, nws);

  dim3 grid(KBLOCKS, B_);
  iattn_main<<<grid, THREADS, 0, stream>>>(qry, key, val, attn, norm, acc);

  iattn_finalize<<<(B_ * Q_ * D_ + 255) / 256, 256, 0, stream>>>(acc, norm, out);
}